// A2EvULoss_78228534330114
// MI455X (gfx1250) — compile-verified
//
#include <hip/hip_runtime.h>
#include <math.h>

typedef __attribute__((ext_vector_type(2))) float v2f;
typedef __attribute__((ext_vector_type(8))) float v8f;

#define WS_BINS 96  // 84 bins (21 thresholds x 4 cols) + 2 totals + pad

static __device__ __forceinline__ float fast_tanh(float x) {
#if __has_builtin(__builtin_amdgcn_tanhf)
  return __builtin_amdgcn_tanhf(x);   // v_tanh_f32 (CDNA5 TRANS op)
#else
  float r;
  asm volatile("v_tanh_f32 %0, %1" : "=v"(r) : "v"(x));
  return r;
#endif
}

// ---------------------------------------------------------------------------
// Kernel 0: init workspace scalars (minbits=+inf, maxbits=0, bins=0)
// ---------------------------------------------------------------------------
__global__ void init_ws(unsigned* __restrict__ minmax, float* __restrict__ bins) {
  int t = threadIdx.x;
  if (t == 0) minmax[0] = 0x7F800000u;  // +inf bit pattern (unc > 0 always)
  if (t == 1) minmax[1] = 0x00000000u;
  if (t < WS_BINS) bins[t] = 0.0f;
}

// ---------------------------------------------------------------------------
// Kernel 1: per-row reduction over C. One thread per row, float4 loads.
//   evidence = exp(x); alpha = evidence+1
//   unc = C / sum(alpha); max_alpha; correct = (target == argmax)
// Writes one packed float4 {unc, max_alpha, correct, 0} per row so kernel 2
// needs a single vector load per element. Wave-shuffle min/max of unc, then
// 2 atomics per wave on uint bit patterns.
// ---------------------------------------------------------------------------
__global__ void row_stats(const float* __restrict__ out,
                          const int*  __restrict__ target,
                          const int*  __restrict__ ncls,
                          float4* __restrict__ pack,
                          unsigned* __restrict__ minmax,
                          int N, int C) {
  int i = blockIdx.x * blockDim.x + threadIdx.x;
  bool valid = (i < N);
  float umin_l = __builtin_inff();
  float umax_l = 0.0f;
  if (valid) {
    const float* row = out + (size_t)i * (size_t)C;
    float mx = -__builtin_inff();
    int arg = 0;
    float s = 0.0f;
    if ((C & 3) == 0) {
      const float4* r4 = (const float4*)row;
      for (int q = 0; q < (C >> 2); ++q) {
        float4 v = r4[q];
        float e0 = __expf(v.x), e1 = __expf(v.y);
        float e2 = __expf(v.z), e3 = __expf(v.w);
        s += (e0 + e1) + (e2 + e3);
        int cb = q << 2;
        if (e0 > mx) { mx = e0; arg = cb; }
        if (e1 > mx) { mx = e1; arg = cb + 1; }
        if (e2 > mx) { mx = e2; arg = cb + 2; }
        if (e3 > mx) { mx = e3; arg = cb + 3; }
      }
    } else {
      for (int c = 0; c < C; ++c) {
        float e = __expf(row[c]);
        s += e;
        if (e > mx) { mx = e; arg = c; }
      }
    }
    float sum_alpha = s + (float)C;           // sum(evidence) + C*1
    float u = (float)(*ncls) / sum_alpha;
    float co = (target[i] == arg) ? 1.0f : 0.0f;
    pack[i] = make_float4(u, mx + 1.0f, co, 0.0f);
    umin_l = u; umax_l = u;
  }
  // wave32 min/max reduction
  for (int off = 16; off > 0; off >>= 1) {
    umin_l = fminf(umin_l, __shfl_down(umin_l, off, 32));
    umax_l = fmaxf(umax_l, __shfl_down(umax_l, off, 32));
  }
  if ((threadIdx.x & 31) == 0) {
    atomicMin(&minmax[0], __float_as_uint(umin_l));
    atomicMax(&minmax[1], __float_as_uint(umax_l));
  }
}

// ---------------------------------------------------------------------------
// Kernel 2: the [21 x N] @ [N x 4] matmul via V_WMMA_F32_16X16X4_F32.
//   M = threshold row (acc0: rows 0..15, acc1: rows 16..31; rows >20 -> +inf)
//   N = weight column (0:ac 1:au 2:ic 3:iu; cols 4..15 masked to zero)
//   K = elements, 4 per WMMA.
// A layout (16x4 f32): lanes 0..15 hold rows M=lane, K=0,1; lanes 16..31 K=2,3.
// B layout (4x16 f32): lanes 0..15 hold N=lane, K=0,1; lanes 16..31 K=2,3.
// n_au/n_iu use (1-low) = totals - low@w, so totals accumulated alongside.
// Fully branch-free hot loop: index clamp for the global load, column-index
// clamp + multiplicative mask for the LDS weight fetch (no saveexec).
// ---------------------------------------------------------------------------
__global__ void bin_wmma(const float4* __restrict__ pack,
                         const unsigned* __restrict__ minmax,
                         float* __restrict__ g_bins,
                         int N) {
  __shared__ float s_unc[256];
  __shared__ float s_w[256 * 4];
  __shared__ float s_acc[WS_BINS];

  int tid = threadIdx.x;
  if (tid < WS_BINS) s_acc[tid] = 0.0f;

  float umin = __uint_as_float(minmax[0]);
  float umax = __uint_as_float(minmax[1]);
  float du = umax - umin;

  int lane  = tid & 31;
  int wslot = tid >> 5;           // wave slot 0..7
  int half  = lane >> 4;          // 0 or 1
  int koff  = half * 2;           // K base supplied by this lane
  int colr  = lane & 15;          // A row index / B col index
  int csel  = colr & 3;           // clamped weight column (always-valid load)
  float colmask = (colr < 4) ? 1.0f : 0.0f;
  int row_lo = colr;              // threshold rows 0..15 (acc0)
  int row_hi = colr + 16;         // threshold rows 16..31 (acc1)
  float thr0 = umin + (float)row_lo * 0.05f * du;
  float thr1 = (row_hi <= 20) ? (umin + (float)row_hi * 0.05f * du)
                              : __builtin_inff();

  v8f acc0 = {};
  v8f acc1 = {};
  float tot_au = 0.0f, tot_iu = 0.0f;

  int stride = gridDim.x * 256;
  for (int base = blockIdx.x * 256; base < N; base += stride) {
    int idx = base + tid;
    bool valid = (idx < N);
    int idxc = valid ? idx : (N - 1);      // clamp: load is unconditional
    float4 p = pack[idxc];
    float u  = valid ? p.x : __builtin_inff();
    float ma = p.y;
    float co = p.z;
    float t   = fast_tanh(u);              // v_tanh_f32
    float omt = 1.0f - t;
    float ima = 1.0f - ma;
    float vmask = valid ? 1.0f : 0.0f;
    float ac = vmask * co * ma * omt;
    float au = vmask * co * ma * t;
    float ic = vmask * (1.0f - co) * ima * omt;
    float iu = vmask * (1.0f - co) * ima * t;
    tot_au += au;
    tot_iu += iu;

    __syncthreads();              // protect LDS reuse from previous iteration
    s_unc[tid] = u;
    s_w[tid * 4 + 0] = ac;        // ds_store_b128 (merged)
    s_w[tid * 4 + 1] = au;
    s_w[tid * 4 + 2] = ic;
    s_w[tid * 4 + 3] = iu;
    __syncthreads();

    int wb = wslot * 32;
#pragma unroll
    for (int j = 0; j < 8; ++j) {
      int i0 = wb + 4 * j + koff;               // even -> 8B aligned
      float2 up = *(const float2*)&s_unc[i0];   // one ds_load_b64
      float w0 = s_w[i0 * 4 + csel];            // unconditional ds_load_b32
      float w1 = s_w[(i0 + 1) * 4 + csel];
      v2f a0, a1, bb;
      a0.x = (up.x <= thr0) ? 1.0f : 0.0f;      // low mask, rows 0..15
      a0.y = (up.y <= thr0) ? 1.0f : 0.0f;
      a1.x = (up.x <= thr1) ? 1.0f : 0.0f;      // low mask, rows 16..31
      a1.y = (up.y <= thr1) ? 1.0f : 0.0f;
      bb.x = w0 * colmask;                      // zero cols 4..15
      bb.y = w1 * colmask;
      acc0 = __builtin_amdgcn_wmma_f32_16x16x4_f32(
          false, a0, false, bb, (short)0, acc0, false, false);
      acc1 = __builtin_amdgcn_wmma_f32_16x16x4_f32(
          false, a1, false, bb, (short)0, acc1, false, false);
    }
  }
  __syncthreads();

  // Drain C/D tiles: lane holds D[M][Ncol] with M = r + half*8, Ncol = colr.
  if (colr < 4) {
#pragma unroll
    for (int r = 0; r < 8; ++r) {
      int row0 = r + half * 8;
      atomicAdd(&s_acc[row0 * 4 + colr], acc0[r]);     // ds_add_f32
      int row1 = 16 + row0;
      if (row1 < 21) atomicAdd(&s_acc[row1 * 4 + colr], acc1[r]);
    }
  }
  atomicAdd(&s_acc[84], tot_au);
  atomicAdd(&s_acc[85], tot_iu);
  __syncthreads();
  if (tid < 86) atomicAdd(&g_bins[tid], s_acc[tid]);
}

// ---------------------------------------------------------------------------
// Kernel 3: evu, trapezoidal AUC (dx = 0.05), loss = -log(auc + eps)
// ---------------------------------------------------------------------------
__global__ void finalize(const float* __restrict__ g_bins,
                         float* __restrict__ out) {
  if (threadIdx.x == 0 && blockIdx.x == 0) {
    float totau = g_bins[84];
    float totiu = g_bins[85];
    float auc = 0.0f;
    float prev = 0.0f;
    for (int k = 0; k <= 20; ++k) {
      float nac = g_bins[k * 4 + 0];
      float nau = totau - g_bins[k * 4 + 1];
      float nic = g_bins[k * 4 + 2];
      float niu = totiu - g_bins[k * 4 + 3];
      float evu = (nac + niu) / (nac + nau + nic + niu + 1e-10f);
      if (k > 0) auc += 0.5f * (evu + prev) * 0.05f;
      prev = evu;
    }
    out[0] = -logf(auc + 1e-10f);
  }
}

// ---------------------------------------------------------------------------
extern "C" void kernel_launch(void* const* d_in, const int* in_sizes, int n_in,
                              void* d_out, int out_size, void* d_ws,
                              size_t ws_size, hipStream_t stream) {
  const float* output = (const float*)d_in[0];
  const int*   target = (const int*)d_in[1];
  const int*   ncls   = (const int*)d_in[2];
  int N = in_sizes[1];
  int C = in_sizes[0] / N;

  // workspace layout: pack[N] (float4) | minmax[2] | bins[96]
  float4*   w_pack   = (float4*)d_ws;
  unsigned* w_minmax = (unsigned*)(w_pack + N);
  float*    w_bins   = (float*)(w_minmax + 2);

  init_ws<<<1, 128, 0, stream>>>(w_minmax, w_bins);

  int blocks1 = (N + 255) / 256;
  row_stats<<<blocks1, 256, 0, stream>>>(output, target, ncls, w_pack,
                                         w_minmax, N, C);

  int blocks2 = 512;
  bin_wmma<<<blocks2, 256, 0, stream>>>(w_pack, w_minmax, w_bins, N);

  finalize<<<1, 32, 0, stream>>>(w_bins, (float*)d_out);
}